// ContentUnit_56118042689956
// MI455X (gfx1250) — compile-verified
//
#include <hip/hip_runtime.h>
#include <math.h>

// ---------------------------------------------------------------------------
// Problem constants (match the JAX reference)
// ---------------------------------------------------------------------------
constexpr int B_  = 8, T1_ = 32, T2_ = 32, T3_ = 16, D_ = 512, L_ = 20, DL_ = 128;
constexpr int M_  = B_ * T1_ * T2_ * T3_;          // 131072 flattened positions
constexpr float INV_SQRT_DL = 0.08838834764831845f; // 1/sqrt(128)

typedef __attribute__((ext_vector_type(2))) float v2f;
typedef __attribute__((ext_vector_type(8))) float v8f;

// ---------------------------------------------------------------------------
// CDNA5 f32 WMMA: D(16x16,f32) = A(16x4,f32) * B(4x16,f32) + C
// ---------------------------------------------------------------------------
static __device__ __forceinline__ v8f wmma4(v2f a, v2f b, v8f c) {
  return __builtin_amdgcn_wmma_f32_16x16x4_f32(false, a, false, b, (short)0, c,
                                               false, false);
}

static __device__ __forceinline__ float rowmax16(float v) {
  v = fmaxf(v, __shfl_xor(v, 1, 32));
  v = fmaxf(v, __shfl_xor(v, 2, 32));
  v = fmaxf(v, __shfl_xor(v, 4, 32));
  v = fmaxf(v, __shfl_xor(v, 8, 32));
  return v;
}
static __device__ __forceinline__ float rowsum16(float v) {
  v += __shfl_xor(v, 1, 32);
  v += __shfl_xor(v, 2, 32);
  v += __shfl_xor(v, 4, 32);
  v += __shfl_xor(v, 8, 32);
  return v;
}

// ---------------------------------------------------------------------------
// K1a: f_w_hat = f_w @ Ww^T + bw  (160 rows), f_s_hat = f_s @ Ws^T + bs (8 rows)
// ---------------------------------------------------------------------------
__global__ void __launch_bounds__(128, 1)
k_small_proj(const float* __restrict__ f_w, const float* __restrict__ f_s,
             const float* __restrict__ Ww,  const float* __restrict__ bw,
             const float* __restrict__ Ws,  const float* __restrict__ bs,
             float* __restrict__ fwh, float* __restrict__ fsh) {
  const int t = threadIdx.x;           // output channel 0..127
  const int blk = blockIdx.x;
  if (blk < B_ * L_) {
    const float* src = f_w + (long)blk * D_;
    float acc = bw[t];
    for (int d = 0; d < D_; ++d) acc += src[d] * Ww[t * D_ + d];
    fwh[(long)blk * DL_ + t] = acc;
  } else {
    const int b = blk - B_ * L_;
    const float* src = f_s + (long)b * D_;
    float acc = bs[t];
    for (int d = 0; d < D_; ++d) acc += src[d] * Ws[t * D_ + d];
    fsh[(long)b * DL_ + t] = acc;
  }
}

// K1b: k = f_w_hat @ Wk^T + bk   (160 rows x 128)
__global__ void __launch_bounds__(128, 1)
k_keys(const float* __restrict__ fwh, const float* __restrict__ Wk,
       const float* __restrict__ bk, float* __restrict__ kbuf) {
  const int t = threadIdx.x;
  const int row = blockIdx.x;          // 0..159
  const float* src = fwh + (long)row * DL_;
  float acc = bk[t];
  for (int j = 0; j < DL_; ++j) acc += src[j] * Wk[t * DL_ + j];
  kbuf[(long)row * DL_ + t] = acc;
}

// ---------------------------------------------------------------------------
// K2: f_c_hat[M,128] = f_c[M,512] @ Wc_down^T + bc_down   (WMMA, fp32)
// One wave per 16(M) x 128(N).  Software-pipelined: prefetch A/B for k+4 into
// distinct registers while the 8 WMMAs for k execute.  __launch_bounds__ lifts
// the VGPR budget so the double buffers stay in registers (no scratch spills).
// ---------------------------------------------------------------------------
__global__ void __launch_bounds__(128, 1)
k_gemm_down(const float* __restrict__ f_c, const float* __restrict__ Wc_down,
            const float* __restrict__ bc_down, float* __restrict__ fch) {
  const int wave = threadIdx.x >> 5;
  const int lane = threadIdx.x & 31;
  const int r = lane & 15, h = lane >> 4;
  const long m0 = ((long)blockIdx.x * 4 + wave) * 16;

  const v8f z = {0.f,0.f,0.f,0.f,0.f,0.f,0.f,0.f};
  v8f acc[8];
#pragma unroll
  for (int j = 0; j < 8; ++j) acc[j] = z;

  const float* arow = f_c + (m0 + r) * (long)D_ + 2 * h;   // A: row m0+r
  const float* brow = Wc_down + (long)r * D_ + 2 * h;      // B row n = j*16+r

  v2f a_cur = *(const v2f*)(arow);
  v2f b_cur[8];
#pragma unroll
  for (int j = 0; j < 8; ++j) b_cur[j] = *(const v2f*)(brow + (long)j * 16 * D_);

  for (int kk = 0; kk < D_ - 4; kk += 4) {
    v2f a_nxt = *(const v2f*)(arow + kk + 4);
    v2f b_nxt[8];
#pragma unroll
    for (int j = 0; j < 8; ++j)
      b_nxt[j] = *(const v2f*)(brow + (long)j * 16 * D_ + kk + 4);
#pragma unroll
    for (int j = 0; j < 8; ++j) acc[j] = wmma4(a_cur, b_cur[j], acc[j]);
    a_cur = a_nxt;
#pragma unroll
    for (int j = 0; j < 8; ++j) b_cur[j] = b_nxt[j];
  }
#pragma unroll
  for (int j = 0; j < 8; ++j) acc[j] = wmma4(a_cur, b_cur[j], acc[j]);

#pragma unroll
  for (int j = 0; j < 8; ++j) {
    const int n = j * 16 + r;
    const float bias = bc_down[n];
#pragma unroll
    for (int e = 0; e < 8; ++e) {
      const long row = m0 + e + 8 * h;             // C layout: VGPR e -> row e / e+8
      fch[row * DL_ + n] = acc[j][e] + bias;
    }
  }
}

// ---------------------------------------------------------------------------
// K3: fused  q = f_c_hat@Wq^T+bq  ->  scores = q@k^T/sqrt(dl)  -> softmax(L=20)
//            -> f_caq = attn@f_w_hat  ->  f_cq = f_c_hat*(f_caq + f_s_hat)
// One wave per 16-row slab; q and attn staged via per-wave LDS (C->A layout).
// ---------------------------------------------------------------------------
__global__ void __launch_bounds__(128, 1)
k_content_attn(const float* __restrict__ fch, const float* __restrict__ Wq,
               const float* __restrict__ bq,  const float* __restrict__ kbuf,
               const float* __restrict__ fwh, const float* __restrict__ fsh,
               float* __restrict__ fcq) {
  __shared__ __align__(16) float qs[4][16 * 128];  // per-wave q tile
  __shared__ __align__(16) float ps[4][16 * 32];   // per-wave attn probs (padded)
  const int wave = threadIdx.x >> 5;
  const int lane = threadIdx.x & 31;
  const int r = lane & 15, h = lane >> 4;
  const long m0 = ((long)blockIdx.x * 4 + wave) * 16;
  const int b = (int)(m0 >> 14);                   // 16384 rows per batch

  const v8f z = {0.f,0.f,0.f,0.f,0.f,0.f,0.f,0.f};

  // ---- q = f_c_hat @ Wq^T + bq  (pipelined) ----
  v8f acc[8];
#pragma unroll
  for (int j = 0; j < 8; ++j) acc[j] = z;
  {
    const float* arow = fch + (m0 + r) * (long)DL_ + 2 * h;
    const float* brow = Wq + (long)r * DL_ + 2 * h;
    v2f a_cur = *(const v2f*)(arow);
    v2f b_cur[8];
#pragma unroll
    for (int j = 0; j < 8; ++j) b_cur[j] = *(const v2f*)(brow + (long)j * 16 * DL_);
    for (int kk = 0; kk < DL_ - 4; kk += 4) {
      v2f a_nxt = *(const v2f*)(arow + kk + 4);
      v2f b_nxt[8];
#pragma unroll
      for (int j = 0; j < 8; ++j)
        b_nxt[j] = *(const v2f*)(brow + (long)j * 16 * DL_ + kk + 4);
#pragma unroll
      for (int j = 0; j < 8; ++j) acc[j] = wmma4(a_cur, b_cur[j], acc[j]);
      a_cur = a_nxt;
#pragma unroll
      for (int j = 0; j < 8; ++j) b_cur[j] = b_nxt[j];
    }
#pragma unroll
    for (int j = 0; j < 8; ++j) acc[j] = wmma4(a_cur, b_cur[j], acc[j]);
  }
#pragma unroll
  for (int j = 0; j < 8; ++j) {
    const float bias = bq[j * 16 + r];
#pragma unroll
    for (int e = 0; e < 8; ++e)
      qs[wave][(e + 8 * h) * 128 + j * 16 + r] = acc[j][e] + bias;
  }

  // ---- scores = q @ k^T (N padded 20->32, pipelined) ----
  v8f s0 = z, s1 = z;
  {
    const float* k0row = kbuf + ((long)b * L_ + r) * DL_ + 2 * h;
    const float* k1row = kbuf + ((long)b * L_ + (r < 4 ? 16 + r : r)) * DL_ + 2 * h;
    v2f a_cur = *(const v2f*)(&qs[wave][r * 128 + 2 * h]);
    v2f b0_cur = *(const v2f*)(k0row);
    v2f b1_cur = *(const v2f*)(k1row);
    if (r >= 4) { b1_cur.x = 0.f; b1_cur.y = 0.f; }
    for (int kk = 0; kk < DL_ - 4; kk += 4) {
      v2f a_nxt  = *(const v2f*)(&qs[wave][r * 128 + kk + 4 + 2 * h]);
      v2f b0_nxt = *(const v2f*)(k0row + kk + 4);
      v2f b1_nxt = *(const v2f*)(k1row + kk + 4);
      if (r >= 4) { b1_nxt.x = 0.f; b1_nxt.y = 0.f; }
      s0 = wmma4(a_cur, b0_cur, s0);
      s1 = wmma4(a_cur, b1_cur, s1);
      a_cur = a_nxt; b0_cur = b0_nxt; b1_cur = b1_nxt;
    }
    s0 = wmma4(a_cur, b0_cur, s0);
    s1 = wmma4(a_cur, b1_cur, s1);
  }

  // ---- row softmax over 20 columns ----
#pragma unroll
  for (int e = 0; e < 8; ++e) {
    const float v0 = s0[e] * INV_SQRT_DL;
    const float v1 = (r < 4) ? s1[e] * INV_SQRT_DL : -3.0e38f;
    const float mx = fmaxf(rowmax16(v0), rowmax16(v1));
    const float e0 = __expf(v0 - mx);
    const float e1 = (r < 4) ? __expf(v1 - mx) : 0.f;
    const float inv = 1.f / (rowsum16(e0) + rowsum16(e1));
    ps[wave][(e + 8 * h) * 32 + r]      = e0 * inv;
    ps[wave][(e + 8 * h) * 32 + 16 + r] = e1 * inv;
  }

  // ---- f_caq = attn @ f_w_hat  (K=20, fully unrolled so loads batch) ----
#pragma unroll
  for (int j = 0; j < 8; ++j) acc[j] = z;
#pragma unroll
  for (int kk = 0; kk < L_; kk += 4) {            // kk = 0,4,8,12,16
    const v2f a = *(const v2f*)(&ps[wave][r * 32 + kk + 2 * h]);
    const int k0 = kk + 2 * h;
    const float* fwrow = fwh + ((long)b * L_ + k0) * DL_;
#pragma unroll
    for (int j = 0; j < 8; ++j) {
      const int n = j * 16 + r;
      v2f bb; bb.x = fwrow[n]; bb.y = fwrow[DL_ + n];
      acc[j] = wmma4(a, bb, acc[j]);
    }
  }

  // ---- gate: f_cq = f_c_hat * (f_caq + f_s_hat) ----
#pragma unroll
  for (int j = 0; j < 8; ++j) {
    const int n = j * 16 + r;
    const float fs_v = fsh[(long)b * DL_ + n];
#pragma unroll
    for (int e = 0; e < 8; ++e) {
      const long row = m0 + e + 8 * h;
      const float fchv = fch[row * DL_ + n];
      fcq[row * DL_ + n] = fchv * (acc[j][e] + fs_v);
    }
  }
}

// ---------------------------------------------------------------------------
// K4: per-(b,x,y) T3 self-attention.  One wave per 16-row (z) slab.
//     S = f_cq f_cq^T / sqrt(dl); P = softmax(S); f_cc_hat = P @ f_c_hat.
// For S = X X^T the A- and B-operand lane loads coincide -> one load feeds both.
// Writes f_cc_hat in place over f_cq (slab-private, race-free).
// ---------------------------------------------------------------------------
__global__ void __launch_bounds__(128, 1)
k_self_attn(const float* __restrict__ fch, float* fcq_cc) {
  __shared__ __align__(16) float ps[4][16 * 16];
  const int wave = threadIdx.x >> 5;
  const int lane = threadIdx.x & 31;
  const int r = lane & 15, h = lane >> 4;
  const long m0 = ((long)blockIdx.x * 4 + wave) * 16;

  const v8f z = {0.f,0.f,0.f,0.f,0.f,0.f,0.f,0.f};

  // ---- S = X X^T  (pipelined) ----
  v8f s = z;
  {
    const float* xrow = fcq_cc + (m0 + r) * (long)DL_ + 2 * h;
    v2f a_cur = *(const v2f*)(xrow);
    for (int kk = 0; kk < DL_ - 4; kk += 4) {
      v2f a_nxt = *(const v2f*)(xrow + kk + 4);
      s = wmma4(a_cur, a_cur, s);
      a_cur = a_nxt;
    }
    s = wmma4(a_cur, a_cur, s);
  }

  // ---- softmax rows (full 16) ----
#pragma unroll
  for (int e = 0; e < 8; ++e) {
    const float v = s[e] * INV_SQRT_DL;
    const float mx = rowmax16(v);
    const float ex = __expf(v - mx);
    const float inv = 1.f / rowsum16(ex);
    ps[wave][(e + 8 * h) * 16 + r] = ex * inv;
  }

  // ---- f_cc_hat = P @ f_c_hat  (K=16, fully unrolled) ----
  v8f acc[8];
#pragma unroll
  for (int j = 0; j < 8; ++j) acc[j] = z;
#pragma unroll
  for (int kk = 0; kk < 16; kk += 4) {
    const v2f a = *(const v2f*)(&ps[wave][r * 16 + kk + 2 * h]);
    const int k0 = kk + 2 * h;
    const float* brow = fch + (m0 + k0) * (long)DL_;
#pragma unroll
    for (int j = 0; j < 8; ++j) {
      const int n = j * 16 + r;
      v2f bb; bb.x = brow[n]; bb.y = brow[DL_ + n];
      acc[j] = wmma4(a, bb, acc[j]);
    }
  }
#pragma unroll
  for (int j = 0; j < 8; ++j) {
    const int n = j * 16 + r;
#pragma unroll
    for (int e = 0; e < 8; ++e) {
      const long row = m0 + e + 8 * h;
      fcq_cc[row * DL_ + n] = acc[j][e];           // in-place over slab's f_cq
    }
  }
}

// ---------------------------------------------------------------------------
// K5: out = f_cc_hat @ Wc_up^T + bc_up + f_c + broadcast(sigmoid(f_m*f_s)*f_m)
// 8 waves/block; wave w covers N columns [w*64, w*64+64) of one 16-row slab.
// ---------------------------------------------------------------------------
__global__ void __launch_bounds__(256, 1)
k_gemm_up(const float* __restrict__ fcc, const float* __restrict__ Wc_up,
          const float* __restrict__ bc_up, const float* __restrict__ f_c,
          const float* __restrict__ f_m, const float* __restrict__ f_s,
          float* __restrict__ out) {
  const int wave = threadIdx.x >> 5;               // 0..7
  const int lane = threadIdx.x & 31;
  const int r = lane & 15, h = lane >> 4;
  const int slab = blockIdx.x;                     // = b*1024 + x*32 + y
  const long m0 = (long)slab * 16;
  const int b = slab >> 10;
  const int n_base = wave * 64;

  const v8f z = {0.f,0.f,0.f,0.f,0.f,0.f,0.f,0.f};
  v8f acc[4];
#pragma unroll
  for (int j = 0; j < 4; ++j) acc[j] = z;

  const float* arow = fcc + (m0 + r) * (long)DL_ + 2 * h;
  const float* brow = Wc_up + (long)(n_base + r) * DL_ + 2 * h; // B[l][d]=Wc_up[d][l]

  v2f a_cur = *(const v2f*)(arow);
  v2f b_cur[4];
#pragma unroll
  for (int j = 0; j < 4; ++j) b_cur[j] = *(const v2f*)(brow + (long)j * 16 * DL_);

  for (int kk = 0; kk < DL_ - 4; kk += 4) {
    v2f a_nxt = *(const v2f*)(arow + kk + 4);
    v2f b_nxt[4];
#pragma unroll
    for (int j = 0; j < 4; ++j)
      b_nxt[j] = *(const v2f*)(brow + (long)j * 16 * DL_ + kk + 4);
#pragma unroll
    for (int j = 0; j < 4; ++j) acc[j] = wmma4(a_cur, b_cur[j], acc[j]);
    a_cur = a_nxt;
#pragma unroll
    for (int j = 0; j < 4; ++j) b_cur[j] = b_nxt[j];
  }
#pragma unroll
  for (int j = 0; j < 4; ++j) acc[j] = wmma4(a_cur, b_cur[j], acc[j]);

#pragma unroll
  for (int j = 0; j < 4; ++j) {
    const int n = n_base + j * 16 + r;
    const float fmv = f_m[(long)slab * D_ + n];
    const float fsv = f_s[(long)b * D_ + n];
    const float fbar = fmv / (1.f + __expf(-fmv * fsv));   // sigmoid(fm*fs)*fm
    const float bias = bc_up[n];
#pragma unroll
    for (int e = 0; e < 8; ++e) {
      const long row = m0 + e + 8 * h;
      out[row * D_ + n] = acc[j][e] + bias + f_c[row * D_ + n] + fbar;
    }
  }
}

// ---------------------------------------------------------------------------
// Host-side orchestration. Workspace (floats):
//   fch   : M*128      = 16,777,216   (f_c_hat)
//   fcqc  : M*128      = 16,777,216   (f_cq, overwritten in place by f_cc_hat)
//   fwh   : B*L*128    = 20,480
//   kbuf  : B*L*128    = 20,480
//   fsh   : B*128      = 1,024
// total ~ 134.4 MB
// ---------------------------------------------------------------------------
extern "C" void kernel_launch(void* const* d_in, const int* in_sizes, int n_in,
                              void* d_out, int out_size, void* d_ws, size_t ws_size,
                              hipStream_t stream) {
  const float* f_c     = (const float*)d_in[0];
  const float* f_w     = (const float*)d_in[1];
  const float* f_s     = (const float*)d_in[2];
  const float* f_m     = (const float*)d_in[3];
  const float* Wc_down = (const float*)d_in[4];
  const float* bc_down = (const float*)d_in[5];
  const float* Ww      = (const float*)d_in[6];
  const float* bw      = (const float*)d_in[7];
  const float* Ws      = (const float*)d_in[8];
  const float* bs      = (const float*)d_in[9];
  const float* Wq      = (const float*)d_in[10];
  const float* bq      = (const float*)d_in[11];
  const float* Wk      = (const float*)d_in[12];
  const float* bk      = (const float*)d_in[13];
  const float* Wc_up   = (const float*)d_in[14];
  const float* bc_up   = (const float*)d_in[15];
  float* out = (float*)d_out;

  float* ws   = (float*)d_ws;
  float* fch  = ws;
  float* fcqc = fch  + (size_t)M_ * DL_;
  float* fwh  = fcqc + (size_t)M_ * DL_;
  float* kbuf = fwh  + (size_t)B_ * L_ * DL_;
  float* fsh  = kbuf + (size_t)B_ * L_ * DL_;

  k_small_proj<<<B_ * L_ + B_, DL_, 0, stream>>>(f_w, f_s, Ww, bw, Ws, bs, fwh, fsh);
  k_keys<<<B_ * L_, DL_, 0, stream>>>(fwh, Wk, bk, kbuf);
  k_gemm_down<<<M_ / 64, 128, 0, stream>>>(f_c, Wc_down, bc_down, fch);
  k_content_attn<<<M_ / 64, 128, 0, stream>>>(fch, Wq, bq, kbuf, fwh, fsh, fcqc);
  k_self_attn<<<M_ / 64, 128, 0, stream>>>(fch, fcqc);
  k_gemm_up<<<M_ / 16, 256, 0, stream>>>(fcqc, Wc_up, bc_up, f_c, f_m, f_s, out);
}